// Conv1d_89215060672787
// MI455X (gfx1250) — compile-verified
//
#include <hip/hip_runtime.h>
#include <cstdint>

#define PI_F 3.14159265358979323846f
#define RHO_LIM_F (1.0f - 1e-6f)

constexpr int Bdim = 8, Ndim = 128, Tdim = 1024;
constexpr int TPB = 256;   // 8 waves (wave32)
constexpr int EPT = 4;     // elements per thread -> 1024 per block

#if defined(__HIP_DEVICE_COMPILE__) && __has_builtin(__builtin_amdgcn_global_load_async_to_lds_b128)
#define USE_ASYNC 1
#else
#define USE_ASYNC 0
#endif

#if USE_ASYNC
typedef int v4i_ __attribute__((vector_size(16)));
typedef __attribute__((address_space(1))) v4i_* gptr_b128;
typedef __attribute__((address_space(3))) v4i_* lptr_b128;
#endif

__global__ __launch_bounds__(TPB) void gp_ntk_conv_kernel(
    const float* __restrict__ k,
    const float* __restrict__ leak_p,
    const float* __restrict__ alpha_p,
    const float* __restrict__ beta_p,
    float* __restrict__ out)
{
    __shared__ float c0s[Tdim + 2];   // conv input 0 with +-1 halo (zeros)
    __shared__ float cns[Tdim + 2];   // conv input 1 with +-1 halo (zeros)
#if USE_ASYNC
    __shared__ float kbuf[TPB * 2 * EPT];  // 8 KB staged (gp,ntk) tile
#endif

    const int tid = threadIdx.x;
    const int row = blockIdx.x;                 // row over (b1,b2,n)
    const int n   = row & (Ndim - 1);
    const int b2  = (row >> 7) & (Bdim - 1);
    const int b1  = row >> 10;
    const int t0  = tid * EPT;

    const size_t rowOff = (size_t)row * (size_t)(Tdim * 2);
    const float* krow   = k + rowOff + (size_t)(t0 * 2);   // 8 contiguous floats (4 gp/ntk pairs)

#if USE_ASYNC
    {
        // Stage this thread's 32 bytes global -> LDS asynchronously (2x b128).
        gptr_b128 g0 = (gptr_b128)(uintptr_t)(krow);
        gptr_b128 g1 = (gptr_b128)((uintptr_t)(krow) + 16u);
        lptr_b128 l0 = (lptr_b128)(unsigned)(uintptr_t)(&kbuf[tid * 8]);
        lptr_b128 l1 = (lptr_b128)((unsigned)(uintptr_t)(&kbuf[tid * 8]) + 16u);
        __builtin_amdgcn_global_load_async_to_lds_b128(g0, l0, 0, 0);
        __builtin_amdgcn_global_load_async_to_lds_b128(g1, l1, 0, 0);
    }
#endif

    // ---- uniform scalars (hot in cache, uniform addresses) ----
    const float a    = fmaxf(leak_p[0], 0.0f);          // clamp_pg(leak)
    const float w0   = fmaxf(alpha_p[0], 0.0f);         // clamp_pg(alpha)
    const float w1   = fmaxf(alpha_p[1], 0.0f);
    const float w2   = fmaxf(alpha_p[2], 0.0f);
    const float bb   = fmaxf(beta_p[0], 0.0f);          // clamp_pg(beta)
    const float one_m = (1.0f - a) * (1.0f - a);
    const float coef  = 1.0f + a * a;
    const float inv2pi = 1.0f / (2.0f * PI_F);
    const bool  fast  = (one_m == 0.0f);                // uniform branch (a == 1)

    // ---- variance rows: v_x[b1][t], v_y[b2][n+t] (tiny, stays in cache) ----
    // diag row base for b: element k[b][b][0][t][0] -> float offset 9*b*N*T*2 + 2*t
    const float4* diag1 = (const float4*)(k + (size_t)(9 * b1) * (size_t)(Ndim * Tdim * 2));
    float4 p0 = diag1[t0 >> 1];
    float4 p1 = diag1[(t0 >> 1) + 1];
    float vx[EPT] = { p0.x, p0.z, p1.x, p1.z };

    const float* diag2 = k + (size_t)(9 * b2) * (size_t)(Ndim * Tdim * 2);
    float vy[EPT];
#pragma unroll
    for (int e = 0; e < EPT; ++e) {
        int nt = n + t0 + e;
        vy[e] = (nt < Tdim) ? diag2[2 * nt] : 0.0f;     // zero-padded v
    }

    // ---- pull in the staged (gp, ntk) tile ----
    float kv[2 * EPT];
#if USE_ASYNC
#if __has_builtin(__builtin_amdgcn_s_wait_asynccnt)
    __builtin_amdgcn_s_wait_asynccnt(0);
#else
    asm volatile("s_wait_asynccnt 0" ::: "memory");
#endif
    {
        float4 q0 = *(const float4*)(&kbuf[tid * 8]);
        float4 q1 = *(const float4*)(&kbuf[tid * 8 + 4]);
        kv[0]=q0.x; kv[1]=q0.y; kv[2]=q0.z; kv[3]=q0.w;
        kv[4]=q1.x; kv[5]=q1.y; kv[6]=q1.z; kv[7]=q1.w;
    }
#else
    {
        float4 q0 = *(const float4*)(krow);
        float4 q1 = *(const float4*)(krow + 4);
        kv[0]=q0.x; kv[1]=q0.y; kv[2]=q0.z; kv[3]=q0.w;
        kv[4]=q1.x; kv[5]=q1.y; kv[6]=q1.z; kv[7]=q1.w;
    }
#endif

    // ---- elementwise arccos-kernel math, results staged to LDS for the conv ----
#pragma unroll
    for (int e = 0; e < EPT; ++e) {
        float gp  = kv[2 * e];
        float ntk = kv[2 * e + 1];
        float sx  = sqrtf(fmaxf(vx[e], 0.0f));
        float sy  = sqrtf(fmaxf(vy[e], 0.0f));
        float sxy = sx * sy;
        float denom = fmaxf(sxy, 1e-12f);
        float rho = fminf(fmaxf(gp / denom, -RHO_LIM_F), RHO_LIM_F);
        float c0, c1;
        if (fast) {
            // one_m == 0: c0 = sxy*rho*coef/2, c1 = coef/2 (skips acos/sqrt)
            c0 = sxy * rho * (0.5f * coef);
            c1 = 0.5f * coef;
        } else {
            float theta = acosf(rho);
            float s     = sqrtf(fmaxf(1.0f - rho * rho, 0.0f));
            float t2    = coef * PI_F - one_m * theta;
            c0 = sxy * inv2pi * fmaf(rho, t2, one_m * s);
            c1 = t2 * inv2pi;
        }
        c0s[1 + t0 + e] = c0;
        cns[1 + t0 + e] = c1 * ntk;
    }
    if (tid == 0) {
        c0s[0] = 0.0f;        cns[0] = 0.0f;
        c0s[Tdim + 1] = 0.0f; cns[Tdim + 1] = 0.0f;
    }
    __syncthreads();

    // ---- 3-tap conv from LDS (cross-correlation, pad=1) + beta, interleaved store ----
    float og[2 * EPT];
#pragma unroll
    for (int e = 0; e < EPT; ++e) {
        int tt = t0 + e;  // halo: c0s[1+t] holds x[t], so x[t-1] = c0s[tt]
        float kg = fmaf(w0, c0s[tt], fmaf(w1, c0s[tt + 1], w2 * c0s[tt + 2]));
        float kn = fmaf(w0, cns[tt], fmaf(w1, cns[tt + 1], w2 * cns[tt + 2])) + kg;
        og[2 * e]     = kg + bb;
        og[2 * e + 1] = kn + bb;
    }
    float* orow = out + rowOff + (size_t)(t0 * 2);
    *(float4*)(orow)     = make_float4(og[0], og[1], og[2], og[3]);
    *(float4*)(orow + 4) = make_float4(og[4], og[5], og[6], og[7]);
}

extern "C" void kernel_launch(void* const* d_in, const int* in_sizes, int n_in,
                              void* d_out, int out_size, void* d_ws, size_t ws_size,
                              hipStream_t stream) {
    const float* k     = (const float*)d_in[0];
    const float* leak  = (const float*)d_in[1];
    const float* alpha = (const float*)d_in[2];
    const float* beta  = (const float*)d_in[3];
    float* out = (float*)d_out;

    dim3 grid(Bdim * Bdim * Ndim);   // 8192 rows of T=1024
    dim3 block(TPB);
    hipLaunchKernelGGL(gp_ntk_conv_kernel, grid, block, 0, stream,
                       k, leak, alpha, beta, out);
}